// CRF_77859167142372
// MI455X (gfx1250) — compile-verified
//
#include <hip/hip_runtime.h>

typedef __attribute__((ext_vector_type(2))) float v2f;
typedef __attribute__((ext_vector_type(8))) float v8f;
typedef __attribute__((ext_vector_type(4))) int   v4i;

#define N_TAGS   32
#define T_STEPS  65536
#define N_BLOCKS 512
#define L_STEPS  (T_STEPS / N_BLOCKS)   /* 128 sequential steps per wave */
#define STOP_IDX 31
#define LOG2E    1.4426950408889634f
#define LN2      0.6931471805599453f
#define CHUNK_F  80                     /* padded chunk stride (floats): bank-conflict-free */

#ifndef __has_builtin
#define __has_builtin(x) 0
#endif

#if __has_builtin(__builtin_amdgcn_global_load_async_to_lds_b128)
#define HAVE_ASYNC_LDS 1
#else
#define HAVE_ASYNC_LDS 0
#endif

typedef __attribute__((address_space(1))) v4i gas_v4i;  // global int4
typedef __attribute__((address_space(3))) v4i las_v4i;  // LDS int4

// Raw v_exp_f32 / v_log_f32 (base-2, no denorm fixup sequences)
__device__ __forceinline__ float exp2_fast(float x) {
#if __has_builtin(__builtin_amdgcn_exp2f)
  return __builtin_amdgcn_exp2f(x);
#else
  return exp2f(x);
#endif
}
__device__ __forceinline__ float log2_fast(float x) {
#if __has_builtin(__builtin_amdgcn_logf)
  return __builtin_amdgcn_logf(x);
#else
  return log2f(x);
#endif
}

// Cross-lane broadcast (wave32): ds_bpermute_b32
__device__ __forceinline__ float lane_bcast(float x, int srcLane) {
  return __int_as_float(__builtin_amdgcn_ds_bpermute(srcLane << 2, __float_as_int(x)));
}

// D = A(16x4 f32) * B(4x16 f32) + C(16x16 f32) -- v_wmma_f32_16x16x4_f32
__device__ __forceinline__ v8f wmma4(v2f a, v2f b, v8f c) {
  return __builtin_amdgcn_wmma_f32_16x16x4_f32(false, a, false, b, (short)0, c, false, false);
}

// Wave-private LDS buffers: compiler-ordering fence only (hardware DS is
// in-order per wave; s_wait_dscnt on load results covers the RAW).
__device__ __forceinline__ void wave_lds_fence() {
  __builtin_amdgcn_fence(__ATOMIC_ACQ_REL, "wavefront");
}

// ---------------------------------------------------------------------------
// Stage 1: each wave builds the log2-semiring transfer matrix of one time
// block. C2 (32x32, log2 domain) lives in WMMA C/D tile layout: tile (I,J),
// VGPR r, lanes 0-15 -> row 16I+r, lanes 16-31 -> row 16I+r+8, col 16J+lane&15.
// E is kept in B-operand-packed LDS layout: chunk (Kc,J,c) holds elements
// (K = 16Kc+4c+2hi+v, N = 16J+mlo) at float offset chunk*CHUNK_F + lane*2 + v,
// so each B operand is a single aligned ds_load_b64.
// ---------------------------------------------------------------------------
__global__ void __launch_bounds__(256) crf_stage1(const float* __restrict__ feats,
                                                  const float* __restrict__ trans,
                                                  float* __restrict__ blockP) {
  __shared__ float sTmax2[N_TAGS];                       // rowmax(trans)*log2e
  __shared__ float sTexp[N_TAGS * N_TAGS];               // exp2(trans2 - rowmax2)
  __shared__ __align__(16) float sE[8][16 * CHUNK_F];    // per-wave packed E (5KB)
  __shared__ __align__(16) float sFeat[8][L_STEPS * N_TAGS]; // staged emissions (16KB/wave)

  const int tid  = threadIdx.x;
  const int lane = tid & 31;
  const int wv   = tid >> 5;
  const int w    = blockIdx.x * 8 + wv;           // time-chunk id, 0..511
  const int mlo  = lane & 15;
  const int hi   = lane >> 4;

  const size_t t0 = (size_t)w * L_STEPS;
  const float* gsrc = feats + t0 * N_TAGS;        // this wave's 4096-float chunk
  float* lfeat = sFeat[wv];

  // Kick off the async DMA of this wave's emissions into LDS first so it
  // overlaps the Texp setup below. 32 x b128 (512B per issue) = 16KB.
#if HAVE_ASYNC_LDS
  #pragma unroll
  for (int i = 0; i < (L_STEPS * N_TAGS) / 128; ++i) {
    __builtin_amdgcn_global_load_async_to_lds_b128(
        (gas_v4i*)(gsrc + i * 128 + lane * 4),
        (las_v4i*)(lfeat + i * 128 + lane * 4),
        0, 0);
  }
#else
  for (int i = 0; i < (L_STEPS * N_TAGS) / 128; ++i) {
    float4 t4 = ((const float4*)gsrc)[i * 32 + lane];
    ((float4*)lfeat)[i * 32 + lane] = t4;
  }
#endif

  // rowmax2 of transitions, then Texp = exp2(trans2 - rowmax2) (workgroup-shared)
  if (tid < N_TAGS) {
    float m = -1e30f;
    #pragma unroll
    for (int k = 0; k < N_TAGS; ++k) m = fmaxf(m, trans[tid * N_TAGS + k]);
    sTmax2[tid] = m * LOG2E;
  }
  __syncthreads();
  for (int e = tid; e < N_TAGS * N_TAGS; e += 256)
    sTexp[e] = exp2_fast(trans[e] * LOG2E - sTmax2[e >> 5]);
  __syncthreads();

  const float tmax2_l = sTmax2[lane];

  // Fixed A operands in WMMA 16x4 A layout:
  // VGPR v, lanes0-15: K = 4c+v, lanes16-31: K = 4c+v+2, M = 16I+(lane&15)
  v2f A[2][2][4];
  #pragma unroll
  for (int I = 0; I < 2; ++I)
    #pragma unroll
    for (int Kc = 0; Kc < 2; ++Kc)
      #pragma unroll
      for (int c = 0; c < 4; ++c) {
        const int M = 16 * I + mlo;
        const int K = 16 * Kc + 4 * c + 2 * hi;
        A[I][Kc][c].x = sTexp[M * N_TAGS + K];
        A[I][Kc][c].y = sTexp[M * N_TAGS + K + 1];
      }

  // Wait for the staged emissions before first use.
#if HAVE_ASYNC_LDS
 #if __has_builtin(__builtin_amdgcn_s_wait_asynccnt)
  __builtin_amdgcn_s_wait_asynccnt(0);
 #else
  asm volatile("s_wait_asynccnt 0" ::: "memory");
 #endif
#endif
  wave_lds_fence();

  // C2 := log2e * (trans[i][j] + emit_{t0}[i])
  v8f C2[2][2];
  {
    const float ev = lfeat[lane];
    #pragma unroll
    for (int I = 0; I < 2; ++I)
      #pragma unroll
      for (int r = 0; r < 8; ++r) {
        const int i = 16 * I + r + 8 * hi;
        const float ei = lane_bcast(ev, i);
        #pragma unroll
        for (int J = 0; J < 2; ++J)
          C2[I][J][r] = (trans[i * N_TAGS + 16 * J + mlo] + ei) * LOG2E;
      }
  }

  float* E = sE[wv];

  for (int s = 1; s < L_STEPS; ++s) {
    const float evs2 = lfeat[s * N_TAGS + lane] * LOG2E;

    // Hoisted row-bias broadcasts: er[I][r] = emit2[i] + tmax2[i], i = 16I+r+8hi.
    // Depends only on evs2, so these bpermutes overlap the cmax/exp/WMMA work.
    float er[2][8];
    #pragma unroll
    for (int I = 0; I < 2; ++I)
      #pragma unroll
      for (int r = 0; r < 8; ++r)
        er[I][r] = lane_bcast(evs2 + tmax2_l, 16 * I + r + 8 * hi);

    // column max of C2 per tile-column (reduce own half-rows, swap halves)
    float cmax[2];
    #pragma unroll
    for (int J = 0; J < 2; ++J) {
      float m = -1e30f;
      #pragma unroll
      for (int I = 0; I < 2; ++I)
        #pragma unroll
        for (int r = 0; r < 8; ++r) m = fmaxf(m, C2[I][J][r]);
      cmax[J] = fmaxf(m, lane_bcast(m, lane ^ 16));
    }

    // E = exp2(C2 - colmax2) -> LDS in packed B-operand layout, b64 stores.
    // Row pair (r=2q, 2q+1) -> chunk (I*2+J)*4 + (q>>1) + 2*hi, pos ((q&1)*16+mlo)*2.
    #pragma unroll
    for (int I = 0; I < 2; ++I)
      #pragma unroll
      for (int J = 0; J < 2; ++J)
        #pragma unroll
        for (int q = 0; q < 4; ++q) {
          v2f p;
          p.x = exp2_fast(C2[I][J][2 * q]     - cmax[J]);
          p.y = exp2_fast(C2[I][J][2 * q + 1] - cmax[J]);
          const int chunk = (I * 2 + J) * 4 + (q >> 1) + 2 * hi;
          const int pos   = chunk * CHUNK_F + (((q & 1) << 4) + mlo) * 2;
          *(v2f*)&E[pos] = p;
        }
    wave_lds_fence();

    // B operands: one aligned ds_load_b64 each, already in WMMA layout.
    v2f B[2][2][4];
    #pragma unroll
    for (int Kc = 0; Kc < 2; ++Kc)
      #pragma unroll
      for (int J = 0; J < 2; ++J)
        #pragma unroll
        for (int c = 0; c < 4; ++c)
          B[Kc][J][c] = *(const v2f*)&E[((Kc * 2 + J) * 4 + c) * CHUNK_F + lane * 2];
    wave_lds_fence();

    // S = Texp x E: two 4-deep WMMA chains per tile (halves matrix-unit latency),
    // then C2' = log2(SA+SB) + cmax2_j + er_i
    #pragma unroll
    for (int I = 0; I < 2; ++I) {
      #pragma unroll
      for (int J = 0; J < 2; ++J) {
        v8f accA = {};
        v8f accB = {};
        #pragma unroll
        for (int c = 0; c < 4; ++c) {
          accA = wmma4(A[I][0][c], B[0][J][c], accA);
          accB = wmma4(A[I][1][c], B[1][J][c], accB);
        }
        #pragma unroll
        for (int r = 0; r < 8; ++r)
          C2[I][J][r] = log2_fast(accA[r] + accB[r]) + cmax[J] + er[I][r];
      }
    }
  }

  // emit block transfer matrix (row-major, log2 domain)
  float* O = blockP + (size_t)w * (N_TAGS * N_TAGS);
  #pragma unroll
  for (int I = 0; I < 2; ++I)
    #pragma unroll
    for (int J = 0; J < 2; ++J)
      #pragma unroll
      for (int r = 0; r < 8; ++r)
        O[(16 * I + r + 8 * hi) * N_TAGS + 16 * J + mlo] = C2[I][J][r];
}

// ---------------------------------------------------------------------------
// Stage 2: pairwise log2-semiring combine: out[p] = src[2p+1] o src[2p].
// One wave per pair; max-shifted exp2, 32 WMMAs, log2.
// ---------------------------------------------------------------------------
__global__ void __launch_bounds__(256) crf_combine(const float* __restrict__ src,
                                                   float* __restrict__ dst, int nPairs) {
  const int wid  = blockIdx.x * 8 + (threadIdx.x >> 5);
  const int lane = threadIdx.x & 31;
  if (wid >= nPairs) return;               // whole-wave uniform exit (EXEC stays full)
  const int mlo = lane & 15;
  const int hi  = lane >> 4;

  const float* PR = src + (size_t)(2 * wid) * 1024;      // earlier factor
  const float* PL = src + (size_t)(2 * wid + 1) * 1024;  // later factor

  float rmax = -1e30f, cmax = -1e30f;                    // rowmax(PL), colmax(PR)
  #pragma unroll
  for (int k = 0; k < 32; ++k) {
    rmax = fmaxf(rmax, PL[lane * 32 + k]);
    cmax = fmaxf(cmax, PR[k * 32 + lane]);
  }

  float rmA[2], cmB[2];
  #pragma unroll
  for (int I = 0; I < 2; ++I) rmA[I] = lane_bcast(rmax, 16 * I + mlo);
  #pragma unroll
  for (int J = 0; J < 2; ++J) cmB[J] = lane_bcast(cmax, 16 * J + mlo);

  v2f Aop[2][2][4], Bop[2][2][4];
  #pragma unroll
  for (int I = 0; I < 2; ++I)
    #pragma unroll
    for (int Kc = 0; Kc < 2; ++Kc)
      #pragma unroll
      for (int c = 0; c < 4; ++c) {
        const int M = 16 * I + mlo;
        const int K = 16 * Kc + 4 * c + 2 * hi;
        Aop[I][Kc][c].x = exp2_fast(PL[M * 32 + K] - rmA[I]);
        Aop[I][Kc][c].y = exp2_fast(PL[M * 32 + K + 1] - rmA[I]);
      }
  #pragma unroll
  for (int Kc = 0; Kc < 2; ++Kc)
    #pragma unroll
    for (int J = 0; J < 2; ++J)
      #pragma unroll
      for (int c = 0; c < 4; ++c) {
        const int K = 16 * Kc + 4 * c + 2 * hi;
        const int N = 16 * J + mlo;
        Bop[Kc][J][c].x = exp2_fast(PR[K * 32 + N] - cmB[J]);
        Bop[Kc][J][c].y = exp2_fast(PR[(K + 1) * 32 + N] - cmB[J]);
      }

  float* O = dst + (size_t)wid * 1024;
  #pragma unroll
  for (int I = 0; I < 2; ++I) {
    float ri[8];
    #pragma unroll
    for (int r = 0; r < 8; ++r) ri[r] = lane_bcast(rmax, 16 * I + r + 8 * hi);
    #pragma unroll
    for (int J = 0; J < 2; ++J) {
      v8f accA = {};
      v8f accB = {};
      #pragma unroll
      for (int c = 0; c < 4; ++c) {
        accA = wmma4(Aop[I][0][c], Bop[0][J][c], accA);
        accB = wmma4(Aop[I][1][c], Bop[1][J][c], accB);
      }
      #pragma unroll
      for (int r = 0; r < 8; ++r)
        O[(16 * I + r + 8 * hi) * 32 + 16 * J + mlo] =
            log2_fast(accA[r] + accB[r]) + ri[r] + cmB[J];
    }
  }
}

// ---------------------------------------------------------------------------
// Stage 3: alpha2 = P2 o (init*log2e); out = LN2 * LSE2(alpha2 + trans2[STOP,:])
// ---------------------------------------------------------------------------
__global__ void crf_finalize(const float* __restrict__ P, const float* __restrict__ trans,
                             const float* __restrict__ initA, float* __restrict__ out) {
  const int lane = threadIdx.x & 31;
  float v[32];
  float m = -1e30f;
  #pragma unroll
  for (int j = 0; j < 32; ++j) {
    v[j] = P[lane * 32 + j] + initA[j] * LOG2E;
    m = fmaxf(m, v[j]);
  }
  float s = 0.f;
  #pragma unroll
  for (int j = 0; j < 32; ++j) s += exp2_fast(v[j] - m);
  const float alpha2 = m + log2_fast(s);
  const float term2  = alpha2 + trans[STOP_IDX * 32 + lane] * LOG2E;

  float wm = term2;
  #pragma unroll
  for (int off = 16; off > 0; off >>= 1) wm = fmaxf(wm, lane_bcast(wm, lane ^ off));
  float se = exp2_fast(term2 - wm);
  #pragma unroll
  for (int off = 16; off > 0; off >>= 1) se += lane_bcast(se, lane ^ off);
  if (lane == 0) out[0] = (wm + log2_fast(se)) * LN2;
}

extern "C" void kernel_launch(void* const* d_in, const int* in_sizes, int n_in,
                              void* d_out, int out_size, void* d_ws, size_t ws_size,
                              hipStream_t stream) {
  (void)in_sizes; (void)n_in; (void)out_size; (void)ws_size;
  const float* feats = (const float*)d_in[0];  // [65536, 32]
  const float* trans = (const float*)d_in[1];  // [32, 32]
  const float* initA = (const float*)d_in[2];  // [32]
  float* out  = (float*)d_out;                 // [1]

  float* bufA = (float*)d_ws;                   // 512 matrices (2 MB)
  float* bufB = bufA + (size_t)N_BLOCKS * 1024; // 256 matrices (1 MB)

  crf_stage1<<<dim3(N_BLOCKS / 8), dim3(256), 0, stream>>>(feats, trans, bufA);

  float* src = bufA;
  float* dst = bufB;
  int n = N_BLOCKS;
  while (n > 1) {
    const int pairs = n >> 1;
    const int grid  = (pairs + 7) / 8;
    crf_combine<<<dim3(grid), dim3(256), 0, stream>>>(src, dst, pairs);
    float* t = src; src = dst; dst = t;
    n = pairs;
  }

  crf_finalize<<<dim3(1), dim3(32), 0, stream>>>(src, trans, initA, out);
}